// HGTAttention_81097572483649
// MI455X (gfx1250) — compile-verified
//
#include <hip/hip_runtime.h>
#include <hip/hip_bf16.h>

typedef __attribute__((ext_vector_type(2))) float v2f;
typedef __attribute__((ext_vector_type(8))) float v8f;

#define H_HEADS 4
#define IN_D    128
#define HID     64
#define BN_EPS  1e-5f
// LDS row stride (words) for staged weights: 72 mod 64 = 8 banks/row, so the
// 2-row offset between the lane halves of a wave shifts the 16-bank span by
// 16 banks -> the two halves hit disjoint banks (conflict-free ds_load).
#define LDSW    72

__device__ __forceinline__ v8f wmma_f32(v2f a, v2f b, v8f c) {
    // D = A(16x4,f32) * B(4x16,f32) + C(16x16,f32)
    return __builtin_amdgcn_wmma_f32_16x16x4_f32(false, a, false, b, (short)0, c, false, false);
}

// Stage a [K x HID] weight matrix from global into LDS with padded stride.
// nf4 = K*HID/4 float4's; blockDim.x == 128, nf4 % 128 == 0 (uniform, no div).
__device__ __forceinline__ void stage_weights(float* sW, const float4* Wg4, int nf4) {
    const int tid = threadIdx.x;
    for (int idx4 = tid; idx4 < nf4; idx4 += 128) {
        const int row = idx4 >> 4;        // HID/4 = 16 float4 per row
        const int c4  = idx4 & 15;
        ((float4*)(sW + row * LDSW))[c4] = Wg4[idx4];
    }
}

// ---------------------------------------------------------------------------
// Z[h] = X @ W[h] + b[h]  (X:[N,IN_D], W:[H,IN_D,HID]) + BN batch stats.
// Block = 128 threads = 4 waves; each wave owns a DIFFERENT 16-row stripe and
// covers all 64 output cols with 4 accumulators -> A is read exactly once.
// W[h] staged in LDS. grid = (ceil(N/64), H).
// ---------------------------------------------------------------------------
__global__ void gemm_xw_stats(const float* __restrict__ X,
                              const float* __restrict__ W,
                              const float* __restrict__ bias,
                              float* __restrict__ Z,
                              float* __restrict__ sum,
                              float* __restrict__ sumsq,
                              int N) {
    __shared__ float sW[IN_D * LDSW];                       // 36 KB
    const int h = blockIdx.y;
    stage_weights(sW, (const float4*)(W + (size_t)h * IN_D * HID), IN_D * HID / 4);
    __syncthreads();

    const int lane = threadIdx.x & 31, wid = threadIdx.x >> 5;
    const int lo = lane & 15, hi = lane >> 4;
    const int m0 = (blockIdx.x * 4 + wid) * 16;

    const int rowA  = m0 + lo;
    const int rowAc = rowA < N ? rowA : N - 1;              // branchless clamp
    const float* __restrict__ Arow = X + (size_t)rowAc * IN_D + 2 * hi;
    const float* __restrict__ sWl  = sW + 2 * hi * LDSW + lo;

    v8f acc[4] = {};
    for (int kk = 0; kk < IN_D; kk += 4) {
        v2f a; a.x = Arow[kk]; a.y = Arow[kk + 1];
        const float* __restrict__ sWk = sWl + kk * LDSW;
        #pragma unroll
        for (int t = 0; t < 4; ++t) {
            v2f b; b.x = sWk[t * 16]; b.y = sWk[LDSW + t * 16];
            acc[t] = wmma_f32(a, b, acc[t]);
        }
    }

    #pragma unroll
    for (int t = 0; t < 4; ++t) {
        const int col = t * 16 + lo;
        const float bv = bias[h * HID + col];
        float s = 0.f, s2 = 0.f;
        #pragma unroll
        for (int r = 0; r < 8; ++r) {
            const int row = m0 + r + 8 * hi;                // C/D: lanes16-31 -> M=r+8
            const float z = acc[t][r] + bv;
            if (row < N) {
                Z[((size_t)h * N + row) * HID + col] = z;
                s += z; s2 += z * z;
            }
        }
        atomicAdd(&sum[h * HID + col], s);
        atomicAdd(&sumsq[h * HID + col], s2);
    }
}

// ---------------------------------------------------------------------------
// Fold BN (training-mode, biased var) into affine y = z*a + c.
// ---------------------------------------------------------------------------
__global__ void bn_finalize(const float* __restrict__ sum,
                            const float* __restrict__ sumsq,
                            const float* __restrict__ g,
                            const float* __restrict__ be,
                            float* __restrict__ bn_a,
                            float* __restrict__ bn_c,
                            int N) {
    const int i = blockIdx.x * blockDim.x + threadIdx.x;
    if (i >= H_HEADS * HID) return;
    const float invN = 1.f / (float)N;
    const float mu   = sum[i] * invN;
    const float var  = fmaxf(sumsq[i] * invN - mu * mu, 0.f);
    const float a    = g[i] * rsqrtf(var + BN_EPS);
    bn_a[i] = a;
    bn_c[i] = be[i] - mu * a;
}

// ---------------------------------------------------------------------------
// Out[h] = relu(Z[h]*a + c) @ W[h] + b[h]  (K = 64).  Single output (q).
// ---------------------------------------------------------------------------
__global__ void gemm_bn_hw(const float* __restrict__ Z,
                           const float* __restrict__ bn_a,
                           const float* __restrict__ bn_c,
                           const float* __restrict__ W,
                           const float* __restrict__ bias,
                           float* __restrict__ Out,
                           int N) {
    __shared__ float sW[HID * LDSW];                        // 18 KB
    const int h = blockIdx.y;
    stage_weights(sW, (const float4*)(W + (size_t)h * HID * HID), HID * HID / 4);
    __syncthreads();

    const int lane = threadIdx.x & 31, wid = threadIdx.x >> 5;
    const int lo = lane & 15, hi = lane >> 4;
    const int m0 = (blockIdx.x * 4 + wid) * 16;

    const int rowA  = m0 + lo;
    const int rowAc = rowA < N ? rowA : N - 1;
    const float* __restrict__ Arow = Z + ((size_t)h * N + rowAc) * HID + 2 * hi;
    const float* __restrict__ aa   = bn_a + h * HID + 2 * hi;
    const float* __restrict__ cc   = bn_c + h * HID + 2 * hi;
    const float* __restrict__ sWl  = sW + 2 * hi * LDSW + lo;

    v8f acc[4] = {};
    for (int kk = 0; kk < HID; kk += 4) {
        v2f a;
        a.x = fmaxf(Arow[kk]     * aa[kk]     + cc[kk],     0.f);
        a.y = fmaxf(Arow[kk + 1] * aa[kk + 1] + cc[kk + 1], 0.f);
        const float* __restrict__ sWk = sWl + kk * LDSW;
        #pragma unroll
        for (int t = 0; t < 4; ++t) {
            v2f b; b.x = sWk[t * 16]; b.y = sWk[LDSW + t * 16];
            acc[t] = wmma_f32(a, b, acc[t]);
        }
    }

    #pragma unroll
    for (int t = 0; t < 4; ++t) {
        const int col = t * 16 + lo;
        const float bv = bias[h * HID + col];
        #pragma unroll
        for (int r = 0; r < 8; ++r) {
            const int row = m0 + r + 8 * hi;
            if (row < N) Out[((size_t)h * N + row) * HID + col] = acc[t][r] + bv;
        }
    }
}

// ---------------------------------------------------------------------------
// Dual-output projection: k = A@Wk+bk, v = A@Wv+bv with A = relu(Z*a+c)
// loaded once (halves the hs traffic).  8 accumulators / wave.
// ---------------------------------------------------------------------------
__global__ void gemm_bn_kv(const float* __restrict__ Z,
                           const float* __restrict__ bn_a,
                           const float* __restrict__ bn_c,
                           const float* __restrict__ Wk,
                           const float* __restrict__ bk,
                           const float* __restrict__ Wv,
                           const float* __restrict__ bv_,
                           float* __restrict__ OutK,
                           float* __restrict__ OutV,
                           int N) {
    __shared__ float sWk[HID * LDSW];                       // 18 KB
    __shared__ float sWv[HID * LDSW];                       // 18 KB
    const int h = blockIdx.y;
    stage_weights(sWk, (const float4*)(Wk + (size_t)h * HID * HID), HID * HID / 4);
    stage_weights(sWv, (const float4*)(Wv + (size_t)h * HID * HID), HID * HID / 4);
    __syncthreads();

    const int lane = threadIdx.x & 31, wid = threadIdx.x >> 5;
    const int lo = lane & 15, hi = lane >> 4;
    const int m0 = (blockIdx.x * 4 + wid) * 16;

    const int rowA  = m0 + lo;
    const int rowAc = rowA < N ? rowA : N - 1;
    const float* __restrict__ Arow = Z + ((size_t)h * N + rowAc) * HID + 2 * hi;
    const float* __restrict__ aa   = bn_a + h * HID + 2 * hi;
    const float* __restrict__ cc   = bn_c + h * HID + 2 * hi;
    const int sOff = 2 * hi * LDSW + lo;

    v8f ak[4] = {};
    v8f av[4] = {};
    for (int kk = 0; kk < HID; kk += 4) {
        v2f a;
        a.x = fmaxf(Arow[kk]     * aa[kk]     + cc[kk],     0.f);
        a.y = fmaxf(Arow[kk + 1] * aa[kk + 1] + cc[kk + 1], 0.f);
        const float* __restrict__ pk = sWk + sOff + kk * LDSW;
        const float* __restrict__ pv = sWv + sOff + kk * LDSW;
        #pragma unroll
        for (int t = 0; t < 4; ++t) {
            v2f b;
            b.x = pk[t * 16]; b.y = pk[LDSW + t * 16];
            ak[t] = wmma_f32(a, b, ak[t]);
            b.x = pv[t * 16]; b.y = pv[LDSW + t * 16];
            av[t] = wmma_f32(a, b, av[t]);
        }
    }

    #pragma unroll
    for (int t = 0; t < 4; ++t) {
        const int col = t * 16 + lo;
        const float bkv = bk[h * HID + col];
        const float bvv = bv_[h * HID + col];
        #pragma unroll
        for (int r = 0; r < 8; ++r) {
            const int row = m0 + r + 8 * hi;
            if (row < N) {
                OutK[((size_t)h * N + row) * HID + col] = ak[t][r] + bkv;
                OutV[((size_t)h * N + row) * HID + col] = av[t][r] + bvv;
            }
        }
    }
}

// ---------------------------------------------------------------------------
// Per-edge attention logit: sigmoid(<q[dst], k[src]> / 8); segment max via
// bitwise atomicMax (valid: sigmoid > 0, mmax zero-initialized).
// ---------------------------------------------------------------------------
__global__ void edge_logit(const float* __restrict__ q,
                           const float* __restrict__ k,
                           const int* __restrict__ esrc,
                           const int* __restrict__ edst,
                           float* __restrict__ elog,
                           float* __restrict__ mmax,
                           int E, int Ns, int Nd) {
    const int e = blockIdx.x * blockDim.x + threadIdx.x;
    const int h = blockIdx.y;
    if (e >= E) return;
    const int s = esrc[e], d = edst[e];
    const float4* __restrict__ qr = (const float4*)(q + ((size_t)h * Nd + d) * HID);
    const float4* __restrict__ kr = (const float4*)(k + ((size_t)h * Ns + s) * HID);
    float acc = 0.f;
    #pragma unroll
    for (int i = 0; i < HID / 4; ++i) {
        const float4 a = qr[i], b = kr[i];
        acc += a.x * b.x + a.y * b.y + a.z * b.z + a.w * b.w;
    }
    const float lg = 1.f / (1.f + expf(-acc * 0.125f));   // 1/sqrt(64)
    elog[(size_t)h * E + e] = lg;
    atomicMax((unsigned int*)&mmax[(size_t)h * Nd + d], __float_as_uint(lg));
}

// exp(logit - segmax) in place; segment-sum of exponentials into denom.
__global__ void edge_exp(float* __restrict__ elog,
                         const int* __restrict__ edst,
                         const float* __restrict__ mmax,
                         float* __restrict__ denom,
                         int E, int Nd) {
    const int e = blockIdx.x * blockDim.x + threadIdx.x;
    const int h = blockIdx.y;
    if (e >= E) return;
    const int d = edst[e];
    const float ev = expf(elog[(size_t)h * E + e] - mmax[(size_t)h * Nd + d]);
    elog[(size_t)h * E + e] = ev;
    atomicAdd(&denom[(size_t)h * Nd + d], ev);
}

// t[dst] += att * v[src].  One thread handles 16 of 64 columns of one edge;
// target t is L2-resident so the fp32 atomics complete in L2.
__global__ void edge_scatter(const float* __restrict__ elog,
                             const float* __restrict__ denom,
                             const float* __restrict__ v,
                             const int* __restrict__ esrc,
                             const int* __restrict__ edst,
                             float* __restrict__ T,
                             int E, int Ns, int Nd) {
    const int tid = blockIdx.x * blockDim.x + threadIdx.x;
    const int h   = blockIdx.y;
    const int e   = tid >> 2;
    const int seg = tid & 3;
    if (e >= E) return;
    const int s = esrc[e], d = edst[e];
    const float dn  = denom[(size_t)h * Nd + d];
    const float att = elog[(size_t)h * E + e] / (dn > 0.f ? dn : 1.f);
    const float4* __restrict__ vr = (const float4*)(v + ((size_t)h * Ns + s) * HID + seg * 16);
    float* __restrict__ tr = T + ((size_t)h * Nd + d) * HID + seg * 16;
    #pragma unroll
    for (int i = 0; i < 4; ++i) {
        const float4 vv = vr[i];
        atomicAdd(&tr[i * 4 + 0], att * vv.x);
        atomicAdd(&tr[i * 4 + 1], att * vv.y);
        atomicAdd(&tr[i * 4 + 2], att * vv.z);
        atomicAdd(&tr[i * 4 + 3], att * vv.w);
    }
}

// ---------------------------------------------------------------------------
// out = mean_h( (t[h] + relu(hd_pre[h]*a+c)) @ W_a[h] + b_a[h] )
// Heads folded into the K loop; W_a[h] staged in LDS per head iteration.
// ---------------------------------------------------------------------------
__global__ void gemm_final(const float* __restrict__ T,
                           const float* __restrict__ Zd,
                           const float* __restrict__ bn_a,
                           const float* __restrict__ bn_c,
                           const float* __restrict__ Wa,
                           const float* __restrict__ ba,
                           float* __restrict__ out,
                           int N) {
    __shared__ float sW[HID * LDSW];                        // 18 KB, re-staged per head
    const int lane = threadIdx.x & 31, wid = threadIdx.x >> 5;
    const int lo = lane & 15, hi = lane >> 4;
    const int m0 = (blockIdx.x * 4 + wid) * 16;

    const int rowA  = m0 + lo;
    const int rowAc = rowA < N ? rowA : N - 1;
    const int sOff  = 2 * hi * LDSW + lo;

    v8f acc[4] = {};
    for (int h = 0; h < H_HEADS; ++h) {
        stage_weights(sW, (const float4*)(Wa + (size_t)h * HID * HID), HID * HID / 4);
        __syncthreads();

        const float* __restrict__ Trow = T  + ((size_t)h * N + rowAc) * HID + 2 * hi;
        const float* __restrict__ Zrow = Zd + ((size_t)h * N + rowAc) * HID + 2 * hi;
        const float* __restrict__ aa   = bn_a + h * HID + 2 * hi;
        const float* __restrict__ cc   = bn_c + h * HID + 2 * hi;

        for (int kk = 0; kk < HID; kk += 4) {
            v2f a;
            a.x = Trow[kk]     + fmaxf(Zrow[kk]     * aa[kk]     + cc[kk],     0.f);
            a.y = Trow[kk + 1] + fmaxf(Zrow[kk + 1] * aa[kk + 1] + cc[kk + 1], 0.f);
            const float* __restrict__ sWk = sW + sOff + kk * LDSW;
            #pragma unroll
            for (int t = 0; t < 4; ++t) {
                v2f b; b.x = sWk[t * 16]; b.y = sWk[LDSW + t * 16];
                acc[t] = wmma_f32(a, b, acc[t]);
            }
        }
        __syncthreads();                                   // before re-staging
    }

    const float scale = 1.f / (float)H_HEADS;
    #pragma unroll
    for (int t = 0; t < 4; ++t) {
        const int col = t * 16 + lo;
        float bsum = 0.f;
        #pragma unroll
        for (int h = 0; h < H_HEADS; ++h) bsum += ba[h * HID + col];
        #pragma unroll
        for (int r = 0; r < 8; ++r) {
            const int row = m0 + r + 8 * hi;
            if (row < N) out[(size_t)row * HID + col] = (acc[t][r] + bsum) * scale;
        }
    }
}

// ---------------------------------------------------------------------------
extern "C" void kernel_launch(void* const* d_in, const int* in_sizes, int n_in,
                              void* d_out, int out_size, void* d_ws, size_t ws_size,
                              hipStream_t stream) {
    (void)n_in; (void)out_size; (void)ws_size;

    const float* x_src  = (const float*)d_in[0];
    const float* x_dst  = (const float*)d_in[1];
    const int*   esrc   = (const int*)d_in[2];
    const int*   edst   = (const int*)d_in[3];
    const float* W_src  = (const float*)d_in[4];
    const float* b_src  = (const float*)d_in[5];
    const float* g_src  = (const float*)d_in[6];
    const float* be_src = (const float*)d_in[7];
    const float* W_dst  = (const float*)d_in[8];
    const float* b_dst  = (const float*)d_in[9];
    const float* g_dst  = (const float*)d_in[10];
    const float* be_dst = (const float*)d_in[11];
    const float* W_k    = (const float*)d_in[12];
    const float* b_k    = (const float*)d_in[13];
    const float* W_v    = (const float*)d_in[14];
    const float* b_v    = (const float*)d_in[15];
    const float* W_q    = (const float*)d_in[16];
    const float* b_q    = (const float*)d_in[17];
    const float* W_a    = (const float*)d_in[18];
    const float* b_a    = (const float*)d_in[19];

    const int N_src = in_sizes[0] / IN_D;
    const int N_dst = in_sizes[1] / IN_D;
    const int E     = in_sizes[2];

    // ---- workspace carve-out (floats) ----
    float* ws = (float*)d_ws;
    size_t off = 0;
    auto carve = [&](size_t n) { float* p = ws + off; off += n; return p; };
    float* hs_pre = carve((size_t)H_HEADS * N_src * HID);
    float* hd_pre = carve((size_t)H_HEADS * N_dst * HID);
    float* kb     = carve((size_t)H_HEADS * N_src * HID);
    float* vb     = carve((size_t)H_HEADS * N_src * HID);
    float* qb     = carve((size_t)H_HEADS * N_dst * HID);
    float* tb     = carve((size_t)H_HEADS * N_dst * HID);
    float* ebuf   = carve((size_t)H_HEADS * E);
    float* mb     = carve((size_t)H_HEADS * N_dst);
    float* db     = carve((size_t)H_HEADS * N_dst);
    float* sum_s  = carve(H_HEADS * HID);
    float* sq_s   = carve(H_HEADS * HID);
    float* sum_d  = carve(H_HEADS * HID);
    float* sq_d   = carve(H_HEADS * HID);
    float* bn_as  = carve(H_HEADS * HID);
    float* bn_cs  = carve(H_HEADS * HID);
    float* bn_ad  = carve(H_HEADS * HID);
    float* bn_cd  = carve(H_HEADS * HID);

    // ---- zero init (graph-capture safe) ----
    hipMemsetAsync(tb, 0, (size_t)H_HEADS * N_dst * HID * sizeof(float), stream);
    hipMemsetAsync(mb, 0, (size_t)H_HEADS * N_dst * sizeof(float), stream);
    hipMemsetAsync(db, 0, (size_t)H_HEADS * N_dst * sizeof(float), stream);
    hipMemsetAsync(sum_s, 0, (size_t)4 * H_HEADS * HID * sizeof(float), stream); // sum_s..sq_d contiguous

    const dim3 blk(128);                                   // 4 waves, 64 rows/block
    const dim3 grdS((N_src + 63) / 64, H_HEADS);
    const dim3 grdD((N_dst + 63) / 64, H_HEADS);

    // 1) pre-BN linears + batch stats
    gemm_xw_stats<<<grdS, blk, 0, stream>>>(x_src, W_src, b_src, hs_pre, sum_s, sq_s, N_src);
    gemm_xw_stats<<<grdD, blk, 0, stream>>>(x_dst, W_dst, b_dst, hd_pre, sum_d, sq_d, N_dst);

    // 2) fold BN into affine
    bn_finalize<<<1, 256, 0, stream>>>(sum_s, sq_s, g_src, be_src, bn_as, bn_cs, N_src);
    bn_finalize<<<1, 256, 0, stream>>>(sum_d, sq_d, g_dst, be_dst, bn_ad, bn_cd, N_dst);

    // 3) projections: k,v fused (shared A); q single
    gemm_bn_kv<<<grdS, blk, 0, stream>>>(hs_pre, bn_as, bn_cs, W_k, b_k, W_v, b_v, kb, vb, N_src);
    gemm_bn_hw<<<grdD, blk, 0, stream>>>(hd_pre, bn_ad, bn_cd, W_q, b_q, qb, N_dst);

    // 4) edge phase: logits -> segment softmax -> weighted scatter
    const dim3 egrd((E + 255) / 256, H_HEADS);
    edge_logit<<<egrd, 256, 0, stream>>>(qb, kb, esrc, edst, ebuf, mb, E, N_src, N_dst);
    edge_exp<<<egrd, 256, 0, stream>>>(ebuf, edst, mb, db, E, N_dst);
    const dim3 sgrd((unsigned)(((size_t)E * 4 + 255) / 256), H_HEADS);
    edge_scatter<<<sgrd, 256, 0, stream>>>(ebuf, db, vb, esrc, edst, tb, E, N_src, N_dst);

    // 5) output projection, mean over heads folded into K loop
    gemm_final<<<dim3((N_dst + 63) / 64, 1), blk, 0, stream>>>(
        tb, hd_pre, bn_ad, bn_cd, W_a, b_a, (float*)d_out, N_dst);
}